// TransNet_56882546868624
// MI455X (gfx1250) — compile-verified
//
#include <hip/hip_runtime.h>
#include <math.h>
#include <stdint.h>

// ---------------------------------------------------------------------------
// Model constants (must match reference)
// ---------------------------------------------------------------------------
constexpr int Bb  = 4;
constexpr int Nn  = 1024;
constexpr int Vv  = 20000;
constexpr int Dd  = 512;
constexpr int Hh  = 8;
constexpr int DHh = 64;
constexpr int Ww  = 512;
constexpr int NBb = 2;
constexpr int NSs = 512;
constexpr int PMm = 128;     // pos-bias MLP width
constexpr int FFf = 2048;
constexpr int HD  = Hh * DHh; // 512
constexpr int ROWS = Bb * Nn; // 4096

typedef __attribute__((ext_vector_type(2))) float v2f;
typedef __attribute__((ext_vector_type(8))) float v8f;

__device__ __forceinline__ v8f wmma4(v2f a, v2f b, v8f c) {
  // D = A(16x4,f32) * B(4x16,f32) + C(16x16,f32)  -- exact fp32 matrix op
  return __builtin_amdgcn_wmma_f32_16x16x4_f32(false, a, false, b, (short)0, c,
                                               false, false);
}

__device__ __forceinline__ float gelu_tanh(float x) {
  float x3 = x * x * x;
  return 0.5f * x * (1.f + tanhf(0.7978845608028654f * (x + 0.044715f * x3)));
}
__device__ __forceinline__ float silu_f(float x) {
  return x / (1.f + __expf(-x));
}

// ---------------------------------------------------------------------------
// DPP16 butterfly reductions within each 16-lane half (VALU only, keeps the
// DS pipe free for the P-tile staging). quad_perm xor1/xor2, then row_ror 4/8
// (rotations are valid butterfly steps for commutative reductions).
// ---------------------------------------------------------------------------
template <int CTRL>
__device__ __forceinline__ float dpp_movf(float x) {
  return __int_as_float(__builtin_amdgcn_update_dpp(
      0, __float_as_int(x), CTRL, 0xF, 0xF, true));
}
__device__ __forceinline__ float red16_max(float x) {
  x = fmaxf(x, dpp_movf<0xB1>(x));    // quad_perm [1,0,3,2]  (xor 1)
  x = fmaxf(x, dpp_movf<0x4E>(x));    // quad_perm [2,3,0,1]  (xor 2)
  x = fmaxf(x, dpp_movf<0x124>(x));   // row_ror:4
  x = fmaxf(x, dpp_movf<0x128>(x));   // row_ror:8
  return x;
}
__device__ __forceinline__ float red16_sum(float x) {
  x += dpp_movf<0xB1>(x);
  x += dpp_movf<0x4E>(x);
  x += dpp_movf<0x124>(x);
  x += dpp_movf<0x128>(x);
  return x;
}

// ---------------------------------------------------------------------------
// CDNA5 async global->LDS copy (ASYNCcnt-tracked, bypasses VGPRs).
// GVS mode: uniform SGPR base, 32-bit per-lane byte offset, LDS byte address
// in the VDST VGPR.
// ---------------------------------------------------------------------------
__device__ __forceinline__ void async_ld16(uint32_t lds_byte, uint32_t off_byte,
                                           const float* base) {
  asm volatile("global_load_async_to_lds_b128 %0, %1, %2 offset:0"
               :: "v"(lds_byte), "v"(off_byte), "s"(base)
               : "memory");
}
__device__ __forceinline__ void wait_async0() {
  asm volatile("s_wait_asynccnt 0x0" ::: "memory");
}
// Wait until at most 4 async ops outstanding: async loads complete in order,
// so with 4 per batch this guarantees the previous batch has fully landed
// while the freshly issued batch stays in flight.
__device__ __forceinline__ void wait_async4() {
  asm volatile("s_wait_asynccnt 0x4" ::: "memory");
}

// ---------------------------------------------------------------------------
// Generic fp32 WMMA GEMM: C[M,N] = epi(A[M,K] @ B[K,N]).
// Block = 256 threads (8 waves). Tile 64x64, BK=32. Wave (wr,wc) owns 16x32.
// Double-buffered LDS tiles filled by async b128 copies; the k-loop keeps the
// next tile's copies in flight (ASYNCcnt<=4 throttle) while WMMAs consume the
// current one. M % 64 == 0, K % 32 == 0; N edge guarded (non-pipelined path).
// ACT: 0=none, 1=gelu.  BIAS: add bias[col].  RESID: +Rs[row,col].
// A-tile LDS row stride padded to 36 floats -> 16 conflict-free bank pairs.
// ---------------------------------------------------------------------------
template <int ACT, int BIAS, int RESID>
__global__ __launch_bounds__(256) void gemm_k(const float* __restrict__ A,
                                              const float* __restrict__ Bw,
                                              const float* __restrict__ bias,
                                              const float* Rs, float* C, int M,
                                              int N, int K) {
  __shared__ float As[2][64 * 36];
  __shared__ float Bs[2][32 * 64];
  const int tid = threadIdx.x;
  const int lane = tid & 31, wv = tid >> 5;
  const int hl = lane >> 4, ln16 = lane & 15;
  const int m0 = blockIdx.y * 64, n0 = blockIdx.x * 64;
  const int wr = (wv & 3) * 16, wc = (wv >> 2) * 32;
  const uint32_t asb[2] = {(uint32_t)(uintptr_t)(&As[0][0]),
                           (uint32_t)(uintptr_t)(&As[1][0])};
  const uint32_t bsb[2] = {(uint32_t)(uintptr_t)(&Bs[0][0]),
                           (uint32_t)(uintptr_t)(&Bs[1][0])};
  const bool fullN = (n0 + 64 <= N);
  v8f acc0 = {}, acc1 = {};

  auto issueA = [&](int buf, int k0) {  // 2 async b128 per thread
    for (int t = tid; t < 512; t += 256) {
      int r = t >> 3, c4 = (t & 7) * 4;
      async_ld16(asb[buf] + (uint32_t)(r * 36 + c4) * 4,
                 ((uint32_t)(m0 + r) * (uint32_t)K + (uint32_t)(k0 + c4)) * 4,
                 A);
    }
  };
  auto issueB = [&](int buf, int k0) {  // 2 async b128 per thread
    for (int t = tid; t < 512; t += 256) {
      int r = t >> 4, c4 = (t & 15) * 4;
      async_ld16(bsb[buf] + (uint32_t)(r * 64 + c4) * 4,
                 ((uint32_t)(k0 + r) * (uint32_t)N + (uint32_t)(n0 + c4)) * 4,
                 Bw);
    }
  };
  auto compute = [&](int buf) {
#pragma unroll
    for (int ks = 0; ks < 8; ks++) {
      v2f af = *(const v2f*)(&As[buf][(wr + ln16) * 36 + 4 * ks + 2 * hl]);
      int krow = 4 * ks + 2 * hl;
      v2f b0, b1;
      b0.x = Bs[buf][krow * 64 + wc + ln16];
      b0.y = Bs[buf][(krow + 1) * 64 + wc + ln16];
      b1.x = Bs[buf][krow * 64 + wc + 16 + ln16];
      b1.y = Bs[buf][(krow + 1) * 64 + wc + 16 + ln16];
      acc0 = wmma4(af, b0, acc0);
      acc1 = wmma4(af, b1, acc1);
    }
  };

  if (fullN) {  // pipelined double-buffered path
    issueA(0, 0);
    issueB(0, 0);
    int cur = 0;
    for (int k0 = 0; k0 < K; k0 += 32) {
      const bool pre = (k0 + 32) < K;
      if (pre) {
        issueA(cur ^ 1, k0 + 32);
        issueB(cur ^ 1, k0 + 32);
        wait_async4();  // previous batch (4 oldest) has landed
      } else {
        wait_async0();
      }
      __syncthreads();
      compute(cur);
      __syncthreads();
      cur ^= 1;
    }
  } else {  // N-edge block: guarded synchronous B loads, zero fill
    for (int k0 = 0; k0 < K; k0 += 32) {
      issueA(0, k0);
      for (int t = tid; t < 512; t += 256) {
        int r = t >> 4, c4 = (t & 15) * 4;
        float4 b4 = {0.f, 0.f, 0.f, 0.f};
        if (n0 + c4 < N)
          b4 = *(const float4*)(Bw + (size_t)(k0 + r) * N + n0 + c4);
        *(float4*)(&Bs[0][r * 64 + c4]) = b4;
      }
      wait_async0();
      __syncthreads();
      compute(0);
      __syncthreads();
    }
  }
#pragma unroll
  for (int nt = 0; nt < 2; nt++) {
    v8f acc = nt ? acc1 : acc0;
    int col = n0 + wc + nt * 16 + ln16;
    if (col < N) {
      float bv = BIAS ? bias[col] : 0.f;
#pragma unroll
      for (int r = 0; r < 8; r++) {
        int row = m0 + wr + r + 8 * hl;
        float v = acc[r] + bv;
        if (ACT == 1) v = gelu_tanh(v);
        if (RESID) v += Rs[(size_t)row * N + col];
        C[(size_t)row * N + col] = v;
      }
    }
  }
}

// ---------------------------------------------------------------------------
// Sliding-window attention, flash-style online softmax, all matmuls via WMMA.
// grid.x = b*2+nb, grid.y = head. 256 threads = 8 waves; each wave streams
// 16 q-rows per iteration (4 iterations -> 512 rows). Softmax reductions use
// DPP16 (VALU); P is staged per-wave in LDS to re-fragment as an A operand.
// ---------------------------------------------------------------------------
__global__ __launch_bounds__(256) void attn_k(
    const float* __restrict__ qb, const float* __restrict__ kb,
    const float* __restrict__ vb, const float* __restrict__ memk,
    const float* __restrict__ memv, const float* __restrict__ table,
    float scale, int is_xl, float* __restrict__ outb) {
  const int bnb = blockIdx.x, h = blockIdx.y;
  const int b = bnb >> 1, nb = bnb & 1;
  const int tid = threadIdx.x, lane = tid & 31, wv = tid >> 5;
  const int hl = lane >> 4, ln16 = lane & 15;
  const int rowbase = b * Nn + nb * Ww;
  const float* qblk = qb + (size_t)rowbase * HD + h * DHh;
  const float* kcur = kb + (size_t)rowbase * HD;
  const float* vcur = vb + (size_t)rowbase * HD;
  const float* kprev;
  const float* vprev;
  int has_mem;
  if (nb == 1) {
    kprev = kb + (size_t)(b * Nn) * HD;
    vprev = vb + (size_t)(b * Nn) * HD;
    has_mem = 1;
  } else if (is_xl) {
    kprev = memk + (size_t)b * Ww * HD;
    vprev = memv + (size_t)b * Ww * HD;
    has_mem = 1;
  } else {
    kprev = kcur;  // dummy; masked out
    vprev = vcur;
    has_mem = 0;
  }
  __shared__ float Pl[8 * 256];
  __shared__ float tb[Ww];  // this head's bias table column
  float* Pw = Pl + wv * 256;
  for (int d = tid; d < Ww; d += 256) tb[d] = table[d * Hh + h];
  __syncthreads();

  for (int it = 0; it < 4; ++it) {
    const int i0 = 16 * (wv + 8 * it);
    v2f qa[16];
#pragma unroll
    for (int s = 0; s < 16; s++)
      qa[s] = *(const v2f*)(qblk + (size_t)(i0 + ln16) * HD + 4 * s + 2 * hl);
    v8f O0 = {}, O1 = {}, O2 = {}, O3 = {};
    float rmax[8], rsum[8];
#pragma unroll
    for (int r = 0; r < 8; r++) {
      rmax[r] = -3.0e38f;
      rsum[r] = 0.f;
    }
    for (int kt = 0; kt < 64; ++kt) {
      const int j0 = 16 * kt;
      const float* Kb_;
      const float* Vb_;
      int jj0;
      if (j0 < Ww) {
        Kb_ = kprev; Vb_ = vprev; jj0 = j0;
      } else {
        Kb_ = kcur; Vb_ = vcur; jj0 = j0 - Ww;
      }
      v8f S = {};
#pragma unroll
      for (int s = 0; s < 16; s++) {
        v2f bf = *(const v2f*)(Kb_ + (size_t)(jj0 + ln16) * HD + h * DHh +
                               4 * s + 2 * hl);
        S = wmma4(qa[s], bf, S);
      }
      const int j = j0 + ln16;
      float p[8], tmax[8];
#pragma unroll
      for (int r = 0; r < 8; r++) {
        int i = i0 + r + 8 * hl;
        bool valid = (j > i) && (j <= i + Ww) && (has_mem || j >= Ww);
        int rel = i + Ww - j;
        rel = rel < 0 ? 0 : (rel > Ww - 1 ? Ww - 1 : rel);
        float sc = valid ? (S[r] * scale + tb[rel]) : -1.0e9f;
        tmax[r] = red16_max(sc);  // DPP row max within 16-lane half
        p[r] = sc;
      }
#pragma unroll
      for (int r = 0; r < 8; r++) {
        float nm = fmaxf(rmax[r], tmax[r]);
        float corr = __expf(rmax[r] - nm);
        rmax[r] = nm;
        float e = __expf(p[r] - nm);
        rsum[r] = rsum[r] * corr + red16_sum(e);
        O0[r] *= corr; O1[r] *= corr; O2[r] *= corr; O3[r] *= corr;
        Pw[(r + 8 * hl) * 16 + ln16] = e;  // stage P row-major for A-frag
      }
#pragma unroll
      for (int s2 = 0; s2 < 4; s2++) {
        v2f pa = *(const v2f*)(Pw + ln16 * 16 + 4 * s2 + 2 * hl);
        int jv = jj0 + 4 * s2 + 2 * hl;
        const float* vr0 = Vb_ + (size_t)jv * HD + h * DHh;
        const float* vr1 = vr0 + HD;
        v2f b0, b1, b2, b3;
        b0.x = vr0[ln16];       b0.y = vr1[ln16];
        b1.x = vr0[16 + ln16];  b1.y = vr1[16 + ln16];
        b2.x = vr0[32 + ln16];  b2.y = vr1[32 + ln16];
        b3.x = vr0[48 + ln16];  b3.y = vr1[48 + ln16];
        O0 = wmma4(pa, b0, O0);
        O1 = wmma4(pa, b1, O1);
        O2 = wmma4(pa, b2, O2);
        O3 = wmma4(pa, b3, O3);
      }
    }
    float* ob = outb + (size_t)rowbase * HD + h * DHh;
#pragma unroll
    for (int r = 0; r < 8; r++) {
      float inv = 1.f / rsum[r];
      float* orow = ob + (size_t)(i0 + r + 8 * hl) * HD;
      orow[ln16]      = O0[r] * inv;
      orow[16 + ln16] = O1[r] * inv;
      orow[32 + ln16] = O2[r] * inv;
      orow[48 + ln16] = O3[r] * inv;
    }
  }
}

// ---------------------------------------------------------------------------
// Wave-per-row LayerNorm over D=512.
// ---------------------------------------------------------------------------
__global__ __launch_bounds__(256) void ln_k(const float* __restrict__ x,
                                            const float* __restrict__ g,
                                            float* __restrict__ y, int rows) {
  int gw = (int)((blockIdx.x * 256 + threadIdx.x) >> 5);
  int lane = threadIdx.x & 31;
  if (gw >= rows) return;
  const float* xr = x + (size_t)gw * Dd;
  float v[16];
  float s = 0.f;
#pragma unroll
  for (int i = 0; i < 16; i++) {
    v[i] = xr[lane + 32 * i];
    s += v[i];
  }
#pragma unroll
  for (int m = 16; m >= 1; m >>= 1) s += __shfl_xor(s, m, 32);
  float mu = s * (1.f / 512.f);
  float q = 0.f;
#pragma unroll
  for (int i = 0; i < 16; i++) {
    float d = v[i] - mu;
    q += d * d;
  }
#pragma unroll
  for (int m = 16; m >= 1; m >>= 1) q += __shfl_xor(q, m, 32);
  float inv = rsqrtf(q * (1.f / 512.f) + 1e-5f);
  float* yr = y + (size_t)gw * Dd;
#pragma unroll
  for (int i = 0; i < 16; i++)
    yr[lane + 32 * i] = (v[i] - mu) * inv * g[lane + 32 * i];
}

// Wave-per-(row,head) l2norm * scale over DH=64 (qk rmsnorm).
__global__ __launch_bounds__(256) void rms_k(float* __restrict__ x,
                                             const float* __restrict__ sc,
                                             int rows) {
  int gw = (int)((blockIdx.x * 256 + threadIdx.x) >> 5);
  int lane = threadIdx.x & 31;
  if (gw >= rows) return;
  float* xr = x + (size_t)gw * DHh;
  float a = xr[lane], b = xr[lane + 32];
  float s = a * a + b * b;
#pragma unroll
  for (int m = 16; m >= 1; m >>= 1) s += __shfl_xor(s, m, 32);
  float inv = rsqrtf(s + 1e-12f);
  xr[lane] = a * inv * sc[lane];
  xr[lane + 32] = b * inv * sc[lane + 32];
}

// Embedding gather: one block per token row.
__global__ void embed_k(const int* __restrict__ x,
                        const float* __restrict__ emb, float* __restrict__ h) {
  int row = blockIdx.x;
  const float* e = emb + (size_t)x[row] * Dd;
  float* hr = h + (size_t)row * Dd;
  for (int i = threadIdx.x; i < Dd; i += blockDim.x) hr[i] = e[i];
}

// Positional-bias table: MLP over scalar distance -> table[W][H].
__global__ __launch_bounds__(128) void postab_k(
    const float* __restrict__ w1, const float* __restrict__ b1,
    const float* __restrict__ w2, const float* __restrict__ b2,
    const float* __restrict__ w3, const float* __restrict__ b3,
    float* __restrict__ table) {
  __shared__ float h1[PMm], h2[PMm];
  int d = blockIdx.x, t = threadIdx.x;
  float dv = (float)d;
  h1[t] = silu_f(dv * w1[t] + b1[t]);
  __syncthreads();
  float acc = b2[t];
  for (int i = 0; i < PMm; i++) acc += h1[i] * w2[i * PMm + t];
  h2[t] = silu_f(acc);
  __syncthreads();
  if (t < Hh) {
    float o = b3[t];
    for (int i = 0; i < PMm; i++) o += h2[i] * w3[i * Hh + t];
    table[d * Hh + t] = o;
  }
}

// ---------------------------------------------------------------------------
// Recurrent layer: tokens cross-attend to state vectors.
// grid (W, H, B), 64 threads. skv row = [s_k(64) | s_v(64)] packed width 128.
// ---------------------------------------------------------------------------
__global__ __launch_bounds__(64) void tok2state_k(
    const float* __restrict__ q2s, const float* __restrict__ skv,
    float* __restrict__ tost, int nb) {
  int w = blockIdx.x, h = blockIdx.y, b = blockIdx.z, t = threadIdx.x;
  __shared__ float q[DHh];
  __shared__ float sc[NSs];
  __shared__ float red[64];
  q[t] = q2s[((size_t)(b * Ww + w)) * HD + h * DHh + t] * 0.125f;
  __syncthreads();
  float lmax = -3e38f;
  for (int j = t; j < NSs; j += 64) {
    const float* kr = skv + ((size_t)(b * NSs + j)) * 128;
    float s = 0.f;
    for (int d = 0; d < DHh; d++) s += q[d] * kr[d];
    sc[j] = s;
    lmax = fmaxf(lmax, s);
  }
  red[t] = lmax;
  __syncthreads();
  for (int m = 32; m >= 1; m >>= 1) {
    if (t < m) red[t] = fmaxf(red[t], red[t + m]);
    __syncthreads();
  }
  float gmax = red[0];
  __syncthreads();
  float lsum = 0.f;
  for (int j = t; j < NSs; j += 64) {
    float e = __expf(sc[j] - gmax);
    sc[j] = e;
    lsum += e;
  }
  red[t] = lsum;
  __syncthreads();
  for (int m = 32; m >= 1; m >>= 1) {
    if (t < m) red[t] += red[t + m];
    __syncthreads();
  }
  float inv = 1.f / red[0];
  float o = 0.f;
  for (int j = 0; j < NSs; j++)
    o += sc[j] * skv[((size_t)(b * NSs + j)) * 128 + 64 + t];
  tost[((size_t)(b * Nn + nb * Ww + w)) * HD + h * DHh + t] = o * inv;
}

// States attend to [states | block tokens]. grid (NS, H, B), 128 threads.
__global__ __launch_bounds__(128) void stateattn_k(
    const float* __restrict__ sq, const float* __restrict__ skv,
    const float* __restrict__ kb, const float* __restrict__ vb,
    float* __restrict__ satt, int nb) {
  int si = blockIdx.x, h = blockIdx.y, b = blockIdx.z, t = threadIdx.x;
  __shared__ float q[DHh];
  __shared__ float sc[NSs + Ww];
  __shared__ float red[128];
  if (t < DHh) q[t] = sq[((size_t)(b * NSs + si)) * HD + h * DHh + t] * 0.125f;
  __syncthreads();
  float lmax = -3e38f;
  for (int j = t; j < NSs + Ww; j += 128) {
    const float* kr =
        (j < NSs) ? (skv + ((size_t)(b * NSs + j)) * 128)
                  : (kb + ((size_t)(b * Nn + nb * Ww + (j - NSs))) * HD +
                     h * DHh);
    float s = 0.f;
    for (int d = 0; d < DHh; d++) s += q[d] * kr[d];
    sc[j] = s;
    lmax = fmaxf(lmax, s);
  }
  red[t] = lmax;
  __syncthreads();
  for (int m = 64; m >= 1; m >>= 1) {
    if (t < m) red[t] = fmaxf(red[t], red[t + m]);
    __syncthreads();
  }
  float gmax = red[0];
  __syncthreads();
  float lsum = 0.f;
  for (int j = t; j < NSs + Ww; j += 128) {
    float e = __expf(sc[j] - gmax);
    sc[j] = e;
    lsum += e;
  }
  red[t] = lsum;
  __syncthreads();
  for (int m = 64; m >= 1; m >>= 1) {
    if (t < m) red[t] += red[t + m];
    __syncthreads();
  }
  float inv = 1.f / red[0];
  __syncthreads();
  if (t < DHh) {
    float o = 0.f;
    for (int j = 0; j < NSs + Ww; j++) {
      const float* vr =
          (j < NSs) ? (skv + ((size_t)(b * NSs + j)) * 128 + 64)
                    : (vb + ((size_t)(b * Nn + nb * Ww + (j - NSs))) * HD +
                       h * DHh);
      o += sc[j] * vr[t];
    }
    satt[((size_t)(b * NSs + si)) * HD + h * DHh + t] = o * inv;
  }
}

// EMA state update: st = sigmoid(beta)*st + (1-sigmoid(beta))*sp
__global__ void ema_k(float* st, const float* __restrict__ sp,
                      const float* __restrict__ beta) {
  size_t i = (size_t)blockIdx.x * 256 + threadIdx.x;
  int d = (int)(i & 511);
  float dec = 1.f / (1.f + __expf(-beta[d]));
  st[i] = dec * st[i] + (1.f - dec) * sp[i];
}

__global__ void copy_k(float* dst, const float* __restrict__ src, size_t n) {
  size_t i = (size_t)blockIdx.x * 256 + threadIdx.x;
  if (i < n) dst[i] = src[i];
}

// Copy block-1 k/v chunk [B][W*HD] into an XL output slot. grid (1024, B).
__global__ void copyxl_k(float* dst, const float* __restrict__ src) {
  int b = blockIdx.y;
  size_t i = (size_t)blockIdx.x * 256 + threadIdx.x;  // < W*HD = 262144
  dst[(size_t)b * Ww * HD + i] = src[((size_t)b * Nn + Ww) * HD + i];
}

// ---------------------------------------------------------------------------
// Workspace layout (floats)
// ---------------------------------------------------------------------------
constexpr size_t S_H = (size_t)ROWS * Dd;  // 2,097,152
constexpr size_t OFF_H = 0;
constexpr size_t OFF_XN = OFF_H + S_H;
constexpr size_t OFF_Q = OFF_XN + S_H;
constexpr size_t OFF_K = OFF_Q + S_H;
constexpr size_t OFF_V = OFF_K + S_H;
constexpr size_t OFF_ATT = OFF_V + S_H;
constexpr size_t OFF_FF1 = OFF_ATT + S_H;
constexpr size_t OFF_TAB = OFF_FF1 + (size_t)ROWS * FFf;
constexpr size_t OFF_ST = OFF_TAB + (size_t)Ww * Hh;
constexpr size_t OFF_SN = OFF_ST + (size_t)Bb * NSs * Dd;
constexpr size_t OFF_SKV = OFF_SN + (size_t)Bb * NSs * Dd;
constexpr size_t OFF_Q2S = OFF_SKV + (size_t)Bb * NSs * 128;
constexpr size_t OFF_SQ = OFF_Q2S + (size_t)Bb * Ww * HD;
constexpr size_t OFF_SATT = OFF_SQ + (size_t)Bb * NSs * HD;
constexpr size_t OFF_SPROJ = OFF_SATT + (size_t)Bb * NSs * HD;
constexpr size_t OFF_TOST = OFF_SPROJ + (size_t)Bb * NSs * Dd;

extern "C" void kernel_launch(void* const* d_in, const int* in_sizes, int n_in,
                              void* d_out, int out_size, void* d_ws,
                              size_t ws_size, hipStream_t stream) {
  (void)in_sizes; (void)n_in; (void)out_size; (void)ws_size;
  const int* x = (const int*)d_in[0];
  const float* tok_emb = (const float*)d_in[1];
  const float* pos_w1 = (const float*)d_in[2];
  const float* pos_b1 = (const float*)d_in[3];
  const float* pos_w2 = (const float*)d_in[4];
  const float* pos_b2 = (const float*)d_in[5];
  const float* pos_w3 = (const float*)d_in[6];
  const float* pos_b3 = (const float*)d_in[7];
  const float* ln_g = (const float*)d_in[8];
  const float* Wq = (const float*)d_in[9];
  const float* Wk = (const float*)d_in[10];
  const float* Wv = (const float*)d_in[11];
  const float* Wo = (const float*)d_in[12];
  const float* q_scale = (const float*)d_in[13];
  const float* k_scale = (const float*)d_in[14];
  const float* Wo_state = (const float*)d_in[15];
  const float* state_norm_g = (const float*)d_in[16];
  const float* Wq2s = (const float*)d_in[17];
  const float* Ws2q = (const float*)d_in[18];
  const float* Ws2kv = (const float*)d_in[19];
  const float* W_state_out = (const float*)d_in[20];
  const float* ema_beta = (const float*)d_in[21];
  const float* ff_g = (const float*)d_in[22];
  const float* ff_w1 = (const float*)d_in[23];
  const float* ff_b1 = (const float*)d_in[24];
  const float* ff_w2 = (const float*)d_in[25];
  const float* ff_b2 = (const float*)d_in[26];
  const float* final_g = (const float*)d_in[27];
  const float* W_logits = (const float*)d_in[28];
  const float* xl_mem_k = (const float*)d_in[29];
  const float* xl_mem_v = (const float*)d_in[30];
  const float* states = (const float*)d_in[31];

  float* ws = (float*)d_ws;
  float* hs = ws + OFF_H;
  float* xn = ws + OFF_XN;
  float* qb = ws + OFF_Q;
  float* kb = ws + OFF_K;
  float* vb = ws + OFF_V;
  float* att = ws + OFF_ATT;
  float* ff1 = ws + OFF_FF1;
  float* table = ws + OFF_TAB;
  float* st = ws + OFF_ST;
  float* sn = ws + OFF_SN;
  float* skv = ws + OFF_SKV;
  float* q2s = ws + OFF_Q2S;
  float* sq = ws + OFF_SQ;
  float* satt = ws + OFF_SATT;
  float* sproj = ws + OFF_SPROJ;
  float* tost = ws + OFF_TOST;

  float* out_logits = (float*)d_out;
  float* out_xlk = out_logits + (size_t)ROWS * Vv;
  float* out_xlv = out_xlk + (size_t)2 * Bb * Ww * HD;
  float* out_states = out_xlv + (size_t)2 * Bb * Ww * HD;

  const dim3 gD(8, 64);     // [4096 x 512] outputs
  const dim3 gFF(32, 64);   // [4096 x 2048]
  const dim3 gS(8, 32);     // [2048 x 512]

  embed_k<<<ROWS, 128, 0, stream>>>(x, tok_emb, hs);
  postab_k<<<Ww, 128, 0, stream>>>(pos_w1, pos_b1, pos_w2, pos_b2, pos_w3,
                                   pos_b3, table);

  for (int l = 0; l < 6; l++) {
    ln_k<<<512, 256, 0, stream>>>(hs, ln_g + (size_t)l * Dd, xn, ROWS);
    gemm_k<0, 0, 0><<<gD, 256, 0, stream>>>(xn, Wq + (size_t)l * Dd * HD,
                                            nullptr, nullptr, qb, ROWS, HD, Dd);
    gemm_k<0, 0, 0><<<gD, 256, 0, stream>>>(xn, Wk + (size_t)l * Dd * HD,
                                            nullptr, nullptr, kb, ROWS, HD, Dd);
    gemm_k<0, 0, 0><<<gD, 256, 0, stream>>>(xn, Wv + (size_t)l * Dd * HD,
                                            nullptr, nullptr, vb, ROWS, HD, Dd);
    const bool qkrms = (l >= 3);
    if (qkrms) {
      rms_k<<<ROWS, 256, 0, stream>>>(qb, q_scale + (size_t)l * DHh,
                                      ROWS * Hh);
      rms_k<<<ROWS, 256, 0, stream>>>(kb, k_scale + (size_t)l * DHh,
                                      ROWS * Hh);
    }
    if (l == 4 || l == 5) {  // save next_xl_{k,v} = block-1 k/v of this layer
      size_t slot = (size_t)(l - 4) * Bb * Ww * HD;
      copyxl_k<<<dim3(1024, Bb), 256, 0, stream>>>(out_xlk + slot, kb);
      copyxl_k<<<dim3(1024, Bb), 256, 0, stream>>>(out_xlv + slot, vb);
    }
    // enhanced recurrence: layer 4 reads xl_mem[1], layer 5 reads xl_mem[0]
    const float* mk = kb;
    const float* mv = vb;
    int is_xl = 0;
    if (l == 4) { mk = xl_mem_k + (size_t)Bb * Ww * HD;
                  mv = xl_mem_v + (size_t)Bb * Ww * HD; is_xl = 1; }
    if (l == 5) { mk = xl_mem_k; mv = xl_mem_v; is_xl = 1; }
    float scale = qkrms ? 8.0f : 0.125f;
    attn_k<<<dim3(Bb * NBb, Hh), 256, 0, stream>>>(qb, kb, vb, mk, mv, table,
                                                   scale, is_xl, att);
    gemm_k<0, 0, 1><<<gD, 256, 0, stream>>>(att, Wo + (size_t)l * HD * Dd,
                                            nullptr, hs, hs, ROWS, Dd, HD);
    if (l == 3) {  // recurrent state scan over the two blocks
      copy_k<<<4096, 256, 0, stream>>>(st, states, (size_t)Bb * NSs * Dd);
      for (int nb = 0; nb < NBb; nb++) {
        ln_k<<<256, 256, 0, stream>>>(st, state_norm_g, sn, Bb * NSs);
        gemm_k<0, 0, 0><<<dim3(2, 32), 256, 0, stream>>>(
            sn, Ws2kv, nullptr, nullptr, skv, Bb * NSs, 128, Dd);
        for (int b = 0; b < Bb; b++)
          gemm_k<0, 0, 0><<<dim3(8, 8), 256, 0, stream>>>(
              xn + ((size_t)b * Nn + (size_t)nb * Ww) * Dd, Wq2s, nullptr,
              nullptr, q2s + (size_t)b * Ww * HD, Ww, HD, Dd);
        tok2state_k<<<dim3(Ww, Hh, Bb), 64, 0, stream>>>(q2s, skv, tost, nb);
        gemm_k<0, 0, 0><<<gS, 256, 0, stream>>>(sn, Ws2q, nullptr, nullptr, sq,
                                                Bb * NSs, HD, Dd);
        stateattn_k<<<dim3(NSs, Hh, Bb), 128, 0, stream>>>(sq, skv, kb, vb,
                                                           satt, nb);
        gemm_k<0, 0, 0><<<gS, 256, 0, stream>>>(satt, W_state_out, nullptr,
                                                nullptr, sproj, Bb * NSs, Dd,
                                                HD);
        ema_k<<<4096, 256, 0, stream>>>(st, sproj, ema_beta);
      }
      gemm_k<0, 0, 1><<<gD, 256, 0, stream>>>(tost, Wo_state, nullptr, hs, hs,
                                              ROWS, Dd, HD);
      copy_k<<<4096, 256, 0, stream>>>(out_states, st, (size_t)Bb * NSs * Dd);
    }
    ln_k<<<512, 256, 0, stream>>>(hs, ff_g + (size_t)l * Dd, xn, ROWS);
    gemm_k<1, 1, 0><<<gFF, 256, 0, stream>>>(
        xn, ff_w1 + (size_t)l * Dd * FFf, ff_b1 + (size_t)l * FFf, nullptr,
        ff1, ROWS, FFf, Dd);
    gemm_k<0, 1, 1><<<gD, 256, 0, stream>>>(
        ff1, ff_w2 + (size_t)l * FFf * Dd, ff_b2 + (size_t)l * Dd, hs, hs,
        ROWS, Dd, FFf);
  }
  ln_k<<<512, 256, 0, stream>>>(hs, final_g, xn, ROWS);
  gemm_k<0, 0, 0><<<dim3((Vv + 63) / 64, 64), 256, 0, stream>>>(
      xn, W_logits, nullptr, nullptr, out_logits, ROWS, Vv, Dd);
}